// Decorrelation_19980187862064
// MI455X (gfx1250) — compile-verified
//
#include <hip/hip_runtime.h>
#include <hip/hip_bf16.h>

// ---------------- problem constants ----------------
#define NSAMP  65536
#define V      10
#define L      45          // V*(V-1)/2
#define NCOEF  14          // DEGREE + ORDER - 1
#define BLK    64          // threads per block == samples per block (2 wave32)
#define NB     (NSAMP / BLK)

// output layout (floats): out | M | log_d | sec | fir | par
#define OUT_OFF   0
#define M_OFF     (NSAMP * V)                 // 655360
#define LOGD_OFF  (M_OFF + NSAMP * V * V)     // 7208960
#define SCAL_OFF  (LOGD_OFF + NSAMP * V)      // 7864320

typedef __attribute__((ext_vector_type(2))) float v2f;
typedef __attribute__((ext_vector_type(8))) float v8f;

typedef int vi4 __attribute__((vector_size(16)));            // int __vector(4)
typedef __attribute__((address_space(1))) int  GInt;         // global int
typedef __attribute__((address_space(3))) int  LInt;         // LDS int
typedef __attribute__((address_space(1))) vi4  GVi4;         // global int4
typedef __attribute__((address_space(3))) vi4  LVi4;         // LDS int4

#if defined(__has_builtin)
#if __has_builtin(__builtin_amdgcn_wmma_f32_16x16x4_f32)
#define HAVE_WMMA_F32_4 1
#endif
#if __has_builtin(__builtin_amdgcn_global_load_async_to_lds_b32) && \
    __has_builtin(__builtin_amdgcn_global_load_async_to_lds_b128)
#define HAVE_ASYNC_BUILTIN 1
#endif
#endif

__device__ __forceinline__ unsigned lds_byte_off(void* p) {
  return (unsigned)(unsigned long long)(__attribute__((address_space(3))) void*)p;
}

__device__ __forceinline__ void async_ld_b32(const void* g, void* l) {
#if defined(HAVE_ASYNC_BUILTIN)
  __builtin_amdgcn_global_load_async_to_lds_b32((GInt*)(void*)g, (LInt*)l, 0, 0);
#else
  asm volatile("global_load_async_to_lds_b32 %0, %1, off"
               :: "v"(lds_byte_off(l)), "v"(g) : "memory");
#endif
}

__device__ __forceinline__ void async_ld_b128(const void* g, void* l) {
#if defined(HAVE_ASYNC_BUILTIN)
  __builtin_amdgcn_global_load_async_to_lds_b128((GVi4*)(void*)g, (LVi4*)l, 0, 0);
#else
  asm volatile("global_load_async_to_lds_b128 %0, %1, off"
               :: "v"(lds_byte_off(l)), "v"(g) : "memory");
#endif
}

__device__ __forceinline__ void wait_async0() {
  asm volatile("s_wait_asynccnt 0x0" ::: "memory");
}

// ======================================================================
// Main kernel: one block = 64 samples.
// ======================================================================
__global__ __launch_bounds__(BLK) void decor_main(
    const float* __restrict__ x, const float* __restrict__ logd,
    const float* __restrict__ params, float* __restrict__ dout) {
  // LDS: 2560 + 3072 + 36864 + 12288 + 2560 = 57344 B
  __shared__ __align__(16) float sX[BLK * V];          // x tile
  __shared__ __align__(16) float sP[48 * 16];          // params transposed [l][k], k padded to 16
  __shared__ __align__(16) float sBasis[BLK * 9 * 16]; // dense basis [n][col c][k]
  __shared__ __align__(16) float sLam[BLK * 48];       // lam [n][l], stride 48
  __shared__ __align__(16) float sOut[BLK * V];

  const int tid  = threadIdx.x;
  const int base = blockIdx.x * BLK;  // first sample of this block

  // ---- zero the padded k=14,15 rows of sP (disjoint from async targets) ----
  for (int e = tid; e < 96; e += BLK) {
    int l = e >> 1, k = 14 + (e & 1);
    sP[l * 16 + k] = 0.f;
  }

  // ---- CDNA5 async global->LDS staging of params (630 f32) and x tile ----
  for (int e = tid; e < NCOEF * L; e += BLK) {
    int k = e / L, l = e - k * L;                // params is [k][l]
    async_ld_b32(params + e, &sP[l * 16 + k]);   // store transposed [l][k]
  }
  for (int e = tid; e < (BLK * V) / 4; e += BLK) // 160 x b128
    async_ld_b128(x + base * V + e * 4, &sX[e * 4]);
  wait_async0();
  __syncthreads();

  // ---- phase 1: dense cubic B-spline basis rows (uniform knots, 4 taps) ----
  {
    const float dkn  = 10.f / 11.f;              // (HI-LO)/(DEGREE-1)
    const float t0   = -5.f - 3.f * dkn;         // first knot
    const float invd = 1.1f;                     // 1/dkn
    const int n = tid;
#pragma unroll
    for (int c = 0; c < 9; ++c) {
      float xv = fminf(fmaxf(sX[n * V + c], -5.f), 5.f);
      float tt = (xv - t0) * invd;
      int s = (int)floorf(tt);
      s = max(3, min(14, s));
      float u = tt - (float)s;
      float um = 1.f - u, u2 = u * u, u3 = u2 * u;
      float w0 = um * um * um * (1.f / 6.f);
      float w1 = (3.f * u3 - 6.f * u2 + 4.f) * (1.f / 6.f);
      float w2 = (-3.f * u3 + 3.f * u2 + 3.f * u + 1.f) * (1.f / 6.f);
      float w3 = u3 * (1.f / 6.f);
      float* dst = &sBasis[(n * 9 + c) * 16];
#pragma unroll
      for (int k = 0; k < 16; k += 4)
        *(float4*)&dst[k] = make_float4(0.f, 0.f, 0.f, 0.f);
      dst[s - 3] = w0; dst[s - 2] = w1; dst[s - 1] = w2; dst[s] = w3; // s<=14 < 16
    }
  }
  __syncthreads();

  // ---- phase 2: lam = basis @ params, per column-group GEMM via WMMA ----
#if defined(HAVE_WMMA_F32_4)
  {
    const int lane = tid & 31, wv = tid >> 5;
    const int m = lane & 15, h = lane >> 4;
#pragma unroll 1
    for (int t = wv * 2; t < wv * 2 + 2; ++t) {  // 2 sample-tiles of 16 per wave
#pragma unroll 1
      for (int c = 0; c < 9; ++c) {
        const int  nn    = m;                    // lambda column within group
        const bool valid = (nn <= 8 - c);
        const int  i     = c + 1 + nn;           // tri row
        const int  lidx  = (i * (i - 1)) / 2 + c;
        const int  lidxc = valid ? lidx : 0;
        v8f acc = {0.f, 0.f, 0.f, 0.f, 0.f, 0.f, 0.f, 0.f};
#pragma unroll
        for (int j = 0; j < 4; ++j) {            // chain K = 16 as 4 x (16x16x4)
          const int kk = 4 * j + 2 * h;
          float2 av = *(const float2*)&sBasis[((t * 16 + m) * 9 + c) * 16 + kk];
          float2 bv = *(const float2*)&sP[lidxc * 16 + kk];
          v2f a; a.x = av.x; a.y = av.y;
          v2f b; b.x = valid ? bv.x : 0.f; b.y = valid ? bv.y : 0.f;
          acc = __builtin_amdgcn_wmma_f32_16x16x4_f32(
              false, a, false, b, (short)0, acc, false, false);
        }
        if (valid) {                             // D[r+8h, n=m] -> lam LDS
#pragma unroll
          for (int r = 0; r < 8; ++r)
            sLam[(t * 16 + r + 8 * h) * 48 + lidx] = acc[r];
        }
      }
    }
  }
#else
  {  // scalar fallback: dense 14-tap dot per (n,l)
    const int n = tid;
    int idx = 0;
    for (int i = 1; i < V; ++i)
      for (int j = 0; j < i; ++j, ++idx) {
        const float* bs = &sBasis[(n * 9 + j) * 16];
        const float* pp = &sP[idx * 16];
        float acc = 0.f;
#pragma unroll
        for (int k = 0; k < NCOEF; ++k) acc = fmaf(bs[k], pp[k], acc);
        sLam[n * 48 + idx] = acc;
      }
  }
#endif
  __syncthreads();

  // ---- phase 3: out = x + lower(lam) * x (per-thread), staged for coalescing ----
  {
    const int n = tid;
    float xr[V];
#pragma unroll
    for (int i = 0; i < V; ++i) xr[i] = sX[n * V + i];
    sOut[n * V + 0] = xr[0];
#pragma unroll
    for (int i = 1; i < V; ++i) {
      float acc = xr[i];
      const int lb = (i * (i - 1)) / 2;
#pragma unroll
      for (int j = 0; j < i; ++j) acc = fmaf(sLam[n * 48 + lb + j], xr[j], acc);
      sOut[n * V + i] = acc;
    }
  }

  // ---- phase 4: stream M (coalesced b128, 1600 float4/block) ----
  {
    float* Mout = dout + M_OFF + base * V * V;
#pragma unroll 1
    for (int it = 0; it < 25; ++it) {
      const int f = (tid + BLK * it) * 4;
      float vals[4];
#pragma unroll
      for (int q = 0; q < 4; ++q) {
        const int ff = f + q;
        const int nl = ff / 100;
        const int r  = ff - nl * 100;
        const int i  = r / 10;
        const int j  = r - i * 10;
        float vv;
        if (j == i)      vv = 1.f;
        else if (j < i)  vv = sLam[nl * 48 + (i * (i - 1)) / 2 + j];
        else             vv = 0.f;
        vals[q] = vv;
      }
      *(float4*)&Mout[f] = make_float4(vals[0], vals[1], vals[2], vals[3]);
    }
  }

  // ---- log_d passthrough (coalesced b128) ----
  {
    const float4* src = (const float4*)(logd + base * V);
    float4* dst = (float4*)(dout + LOGD_OFF + base * V);
    for (int e = tid; e < (BLK * V) / 4; e += BLK) dst[e] = src[e];
  }
  __syncthreads();

  // ---- out store (coalesced b128) ----
  {
    const float4* src = (const float4*)sOut;
    float4* dst = (float4*)(dout + OUT_OFF + base * V);
    for (int e = tid; e < (BLK * V) / 4; e += BLK) dst[e] = src[e];
  }
}

// ======================================================================
// Penalty scalars over the 14x45 params: sec, fir, par
// ======================================================================
__global__ __launch_bounds__(64) void decor_penalty(const float* __restrict__ p,
                                                    float* __restrict__ outs) {
  __shared__ float red[3 * 64];
  const int t = threadIdx.x;
  float par = 0.f, fir = 0.f, sec = 0.f;
  for (int e = t; e < NCOEF * L; e += 64) { float v = p[e]; par = fmaf(v, v, par); }
  for (int e = t; e < (NCOEF - 1) * L; e += 64) {
    float v = p[e + L] - p[e];
    fir = fmaf(v, v, fir);
  }
  for (int e = t; e < (NCOEF - 2) * L; e += 64) {
    float v = p[e + 2 * L] - 2.f * p[e + L] + p[e];
    sec = fmaf(v, v, sec);
  }
  red[t] = sec; red[64 + t] = fir; red[128 + t] = par;
  __syncthreads();
  for (int s = 32; s > 0; s >>= 1) {
    if (t < s) {
      red[t]        += red[t + s];
      red[64 + t]   += red[64 + t + s];
      red[128 + t]  += red[128 + t + s];
    }
    __syncthreads();
  }
  if (t == 0) { outs[0] = red[0]; outs[1] = red[64]; outs[2] = red[128]; }
}

extern "C" void kernel_launch(void* const* d_in, const int* in_sizes, int n_in,
                              void* d_out, int out_size, void* d_ws, size_t ws_size,
                              hipStream_t stream) {
  const float* x      = (const float*)d_in[0];
  const float* logd   = (const float*)d_in[1];
  const float* params = (const float*)d_in[2];
  float* out = (float*)d_out;

  decor_main<<<dim3(NB), dim3(BLK), 0, stream>>>(x, logd, params, out);
  decor_penalty<<<dim3(1), dim3(64), 0, stream>>>(params, out + SCAL_OFF);
}